// MML_54443005444556
// MI455X (gfx1250) — compile-verified
//
#include <hip/hip_runtime.h>
#include <hip/hip_bf16.h>
#include <math.h>

// ---------------------------------------------------------------------------
// CDNA5 (gfx1250) WMMA types
// ---------------------------------------------------------------------------
typedef __attribute__((ext_vector_type(2))) float v2f;
typedef __attribute__((ext_vector_type(8))) float v8f;

#define DCH 64  // node/edge channels

// ---------------------------------------------------------------------------
// Utility kernels
// ---------------------------------------------------------------------------
__global__ void zero_f32_kernel(float* __restrict__ p, long long n) {
    long long i = (long long)blockIdx.x * blockDim.x + threadIdx.x;
    long long stride = (long long)gridDim.x * blockDim.x;
    for (; i < n; i += stride) p[i] = 0.0f;
}

__global__ void deg_count_kernel(const int* __restrict__ dst, float* __restrict__ deg, int E) {
    int i = blockIdx.x * blockDim.x + threadIdx.x;
    int stride = gridDim.x * blockDim.x;
    for (; i < E; i += stride) atomicAdd(deg + dst[i], 1.0f);
}

// ---------------------------------------------------------------------------
// Edge-parallel fused gather + GEMM(K x 64) + relu (+scatter) using
// V_WMMA_F32_16X16X4_F32.
//   MODE 0: message kernel.  A row e = [x[src[e]] , edge_attr[e]]  (K=128)
//           out: atomicAdd into aggr[dst[e]][c]
//   MODE 1: edge update.     A row e = [x[src[e]], x[dst[e]], edge_attr[e]] (K=192)
//           out: store into e_out[e][c]   (row-local -> in-place safe)
// Block = 128 threads = 4 wave32; each wave owns one 16-edge tile per step.
// ---------------------------------------------------------------------------
template <int K, int MODE>
__global__ __launch_bounds__(128)
void edge_gemm_kernel(const float* __restrict__ feat0,  // x, gathered by src
                      const float* __restrict__ feat1,  // x, gathered by dst (MODE 1)
                      const float* __restrict__ efeat,  // edge features [E][64]
                      const int*   __restrict__ src,
                      const int*   __restrict__ dst,
                      const float* __restrict__ W,      // [K][64] row-major
                      const float* __restrict__ bias,   // [64]
                      float*       __restrict__ out,    // MODE0: aggr[N][64]; MODE1: e_out[E][64]
                      int E) {
    constexpr int LDK = K + 4;                 // pad: 16B-aligned rows, conflict-free b64 reads
    __shared__ float wt[64 * LDK];             // W transposed: wt[n][k]
    __shared__ float bs[DCH];
    __shared__ float atile[4][16 * LDK];       // per-wave 16 x K A tile
    __shared__ int   dsti[4][16];

    const int tid  = threadIdx.x;
    const int wave = tid >> 5;
    const int lane = tid & 31;
    const int half = lane >> 4;   // 0 or 1
    const int l16  = lane & 15;

    // --- stage transposed weights + bias into LDS (once per block) ---
    for (int i = tid; i < K * DCH; i += blockDim.x) {
        const int k = i >> 6;          // /64
        const int n = i & 63;
        wt[n * LDK + k] = W[i];
    }
    if (tid < DCH) bs[tid] = bias[tid];
    __syncthreads();

    const int ntiles = (E + 15) >> 4;
    float* at = atile[wave];

    for (int tile = blockIdx.x * 4 + wave; tile < ntiles; tile += gridDim.x * 4) {
        const int e0 = tile << 4;

        // --- destination indices for the scatter epilogue ---
        if (MODE == 0) {
            if (lane < 16) {
                const int e = e0 + lane;
                dsti[wave][lane] = (e < E) ? dst[e] : -1;
            }
        }

        // --- stage A tile: two rows per pass, one float4 per lane per segment ---
        for (int rp = 0; rp < 8; ++rp) {
            const int r = rp * 2 + half;
            const int e = e0 + r;
            float4 v0 = make_float4(0.f, 0.f, 0.f, 0.f);
            float4 v1 = make_float4(0.f, 0.f, 0.f, 0.f);
            float4 v2 = make_float4(0.f, 0.f, 0.f, 0.f);
            if (e < E) {
                const int s = src[e];
                v0 = *reinterpret_cast<const float4*>(feat0 + (size_t)s * DCH + l16 * 4);
                if (MODE == 1) {
                    const int d = dst[e];
                    v1 = *reinterpret_cast<const float4*>(feat1 + (size_t)d * DCH + l16 * 4);
                    v2 = *reinterpret_cast<const float4*>(efeat + (size_t)e * DCH + l16 * 4);
                } else {
                    v1 = *reinterpret_cast<const float4*>(efeat + (size_t)e * DCH + l16 * 4);
                }
            }
            float* row = at + r * LDK;
            *reinterpret_cast<float4*>(row + l16 * 4)       = v0;
            *reinterpret_cast<float4*>(row + 64 + l16 * 4)  = v1;
            if (MODE == 1)
                *reinterpret_cast<float4*>(row + 128 + l16 * 4) = v2;
        }
        // wave-internal cross-lane LDS visibility (CDNA5 split counters)
        asm volatile("s_wait_dscnt 0" ::: "memory");

        // --- accumulators: bias splat (bias depends only on N column) ---
        v8f acc[4];
#pragma unroll
        for (int t = 0; t < 4; ++t) {
            const float bv = bs[t * 16 + l16];
#pragma unroll
            for (int i = 0; i < 8; ++i) acc[t][i] = bv;
        }

        // --- K loop: 16x16x4 fp32 WMMA, 4 output tiles of 16 cols ---
        const float* arow = at + l16 * LDK;
#pragma unroll
        for (int ks = 0; ks < K / 4; ++ks) {
            const int kk = ks * 4 + half * 2;
            const v2f a = *reinterpret_cast<const v2f*>(arow + kk);
#pragma unroll
            for (int t = 0; t < 4; ++t) {
                const v2f b = *reinterpret_cast<const v2f*>(&wt[(t * 16 + l16) * LDK + kk]);
                acc[t] = __builtin_amdgcn_wmma_f32_16x16x4_f32(
                    false, a, false, b, (short)0, acc[t], false, false);
            }
        }

        // --- epilogue: relu, then scatter-add (MODE 0) or store (MODE 1) ---
        if (MODE == 0) {
#pragma unroll
            for (int t = 0; t < 4; ++t) {
#pragma unroll
                for (int i = 0; i < 8; ++i) {
                    float v = fmaxf(acc[t][i], 0.0f);
                    const int row = i + half * 8;          // M index in C/D layout
                    const int d = dsti[wave][row];
                    if (d >= 0) atomicAdd(out + (size_t)d * DCH + t * 16 + l16, v);
                }
            }
        } else {
#pragma unroll
            for (int t = 0; t < 4; ++t) {
#pragma unroll
                for (int i = 0; i < 8; ++i) {
                    float v = fmaxf(acc[t][i], 0.0f);
                    const int e = e0 + i + half * 8;
                    if (e < E) out[(size_t)e * DCH + t * 16 + l16] = v;
                }
            }
        }
    }
}

// ---------------------------------------------------------------------------
// Node update: x = l2norm(relu([aggr/deg , x] @ W + b)).  One wave per node,
// 2 output channels per lane, wave shuffle reduction for the L2 norm.
// ---------------------------------------------------------------------------
__global__ __launch_bounds__(128)
void node_update_kernel(const float* __restrict__ aggr,
                        const float* __restrict__ deg,
                        const float* __restrict__ xin,
                        const float* __restrict__ W,     // [128][64]
                        const float* __restrict__ bias,  // [64]
                        float*       __restrict__ xout,
                        int N) {
    const int wpb  = blockDim.x >> 5;
    const int wave = threadIdx.x >> 5;
    const int lane = threadIdx.x & 31;
    const int c0   = lane * 2;

    for (int n = blockIdx.x * wpb + wave; n < N; n += gridDim.x * wpb) {
        const float inv = 1.0f / fmaxf(deg[n], 1.0f);
        float o0 = bias[c0], o1 = bias[c0 + 1];
        const float* ar = aggr + (size_t)n * DCH;
        const float* xr = xin + (size_t)n * DCH;
#pragma unroll 4
        for (int k = 0; k < DCH; ++k) {
            const float a = ar[k] * inv;
            const float2 w = *reinterpret_cast<const float2*>(W + k * DCH + c0);
            o0 = fmaf(a, w.x, o0);
            o1 = fmaf(a, w.y, o1);
        }
#pragma unroll 4
        for (int k = 0; k < DCH; ++k) {
            const float a = xr[k];
            const float2 w = *reinterpret_cast<const float2*>(W + (DCH + k) * DCH + c0);
            o0 = fmaf(a, w.x, o0);
            o1 = fmaf(a, w.y, o1);
        }
        o0 = fmaxf(o0, 0.0f);
        o1 = fmaxf(o1, 0.0f);
        float ssq = o0 * o0 + o1 * o1;
#pragma unroll
        for (int m = 16; m >= 1; m >>= 1) ssq += __shfl_xor(ssq, m, 32);
        const float scale = 1.0f / fmaxf(sqrtf(ssq), 1e-12f);
        float2 o;
        o.x = o0 * scale;
        o.y = o1 * scale;
        *reinterpret_cast<float2*>(xout + (size_t)n * DCH + c0) = o;
    }
}

// ---------------------------------------------------------------------------
// Host-side orchestration
// ---------------------------------------------------------------------------
extern "C" void kernel_launch(void* const* d_in, const int* in_sizes, int n_in,
                              void* d_out, int out_size, void* d_ws, size_t ws_size,
                              hipStream_t stream) {
    const float* x_in  = (const float*)d_in[0];
    const float* e_in  = (const float*)d_in[1];
    const int*   eidx  = (const int*)d_in[2];
    const float* msgW  = (const float*)d_in[3];
    const float* msgb  = (const float*)d_in[4];
    const float* aggW  = (const float*)d_in[5];
    const float* aggb  = (const float*)d_in[6];
    const float* edgW  = (const float*)d_in[7];
    const float* edgb  = (const float*)d_in[8];

    const int N = in_sizes[0] / DCH;
    const int E = in_sizes[2] / 2;
    const int L = 3;

    const int* src = eidx;
    const int* dst = eidx + E;

    // workspace layout: deg[N] | aggr[N*64] | x_buf[N*64] | e_buf[E*64]
    float* deg  = (float*)d_ws;
    float* aggr = deg + N;
    float* xbuf = aggr + (size_t)N * DCH;
    float* ebuf = xbuf + (size_t)N * DCH;

    float* outp = (float*)d_out;

    const int ZB = 256;
    const int zgrid_deg  = (N + ZB - 1) / ZB;
    const int zgrid_aggr = 4096;
    const int egrid = 4096;   // grid-stride over E/16 = 78125 tiles, 4 waves/block
    const int ngrid = 4096;   // grid-stride over N waves

    // degree (computed once, reused every layer)
    zero_f32_kernel<<<zgrid_deg, ZB, 0, stream>>>(deg, (long long)N);
    deg_count_kernel<<<(E + ZB - 1) / ZB, ZB, 0, stream>>>(dst, deg, E);

    const float* xcur = x_in;
    const float* ecur = e_in;

    for (int l = 0; l < L; ++l) {
        // aggr = 0
        zero_f32_kernel<<<zgrid_aggr, ZB, 0, stream>>>(aggr, (long long)N * DCH);

        // message + scatter-sum (K = 128)
        edge_gemm_kernel<128, 0><<<egrid, 128, 0, stream>>>(
            xcur, nullptr, ecur, src, dst,
            msgW + (size_t)l * 128 * DCH, msgb + (size_t)l * DCH, aggr, E);

        // node update (mean fold + relu + l2norm)
        float* xnext = (l == L - 1) ? outp : xbuf;
        node_update_kernel<<<ngrid, 128, 0, stream>>>(
            aggr, deg, xcur,
            aggW + (size_t)l * 128 * DCH, aggb + (size_t)l * DCH, xnext, N);

        // edge feature update (K = 192), row-local -> in-place on ebuf is safe
        if (l < L - 1) {
            edge_gemm_kernel<192, 1><<<egrid, 128, 0, stream>>>(
                xbuf, xbuf, ecur, src, dst,
                edgW + (size_t)l * 192 * DCH, edgb + (size_t)l * DCH, ebuf, E);
            ecur = ebuf;
        }
        xcur = xbuf;
    }
}